// InterpolationLSTM_59536836657980
// MI455X (gfx1250) — compile-verified
//
#include <hip/hip_runtime.h>

typedef __attribute__((ext_vector_type(2))) float v2f;
typedef __attribute__((ext_vector_type(8))) float v8f;

static inline size_t ws_align(size_t x) { return (x + 255) & ~(size_t)255; }

// Branch-free activations: v_exp_f32 + v_rcp_f32, saturate correctly at +-inf.
__device__ __forceinline__ float sigm(float x) {
  return __builtin_amdgcn_rcpf(1.f + __expf(-x));
}
__device__ __forceinline__ float tanh_fast(float x) {
  return 1.f - 2.f * __builtin_amdgcn_rcpf(__expf(2.f * x) + 1.f);
}

// ---------------------------------------------------------------------------
// flags[0..31] = 0 (convergence chain; must be reset every launch)
__global__ void k_init_flags(int* flags) {
  if (threadIdx.x < 32) flags[threadIdx.x] = 0;
}

// masked = where(mask, x, 0), relayout [T][N][F] -> node-major [N][T*F] (256B/node)
__global__ void k_mask_relayout(const float* __restrict__ x,
                                const unsigned char* __restrict__ mask,
                                float* __restrict__ xw, int N) {
  int n = blockIdx.x * blockDim.x + threadIdx.x;
  int t = blockIdx.y;
  if (n >= N) return;
  float2 v = ((const float2*)x)[(size_t)t * N + n];
  if (!mask[n]) { v.x = 0.f; v.y = 0.f; }
  *(float2*)(xw + (size_t)n * 64 + t * 2) = v;
}

// one wave per node: has_data[n], zero agg/cnt, flags[0] |= any-zero
__global__ void k_prep0(const float* __restrict__ xw, float* __restrict__ agg,
                        float* __restrict__ cnt, int* __restrict__ has_data,
                        int* __restrict__ flags, int N) {
  int tid = blockIdx.x * blockDim.x + threadIdx.x;
  int wave = tid >> 5, lane = tid & 31;
  if (wave >= N) return;
  size_t off = (size_t)wave * 64 + lane * 2;
  float2 v = *(const float2*)(xw + off);
  float2 z = {0.f, 0.f};
  *(float2*)(agg + off) = z;
  int anynz = __any((v.x != 0.f) || (v.y != 0.f));
  int anyz  = __any((v.x == 0.f) || (v.y == 0.f));
  if (lane == 0) {
    cnt[wave] = 0.f;
    has_data[wave] = anynz ? 1 : 0;
    if (anyz) atomicOr(flags, 1);
  }
}

// one wave per edge (32 lanes x float2 = the 256B node block), atomic segment-sum
__global__ void k_scatter(const int* __restrict__ flags, int iter,
                          const int* __restrict__ ei, const int* __restrict__ has_data,
                          const float* __restrict__ xw, float* __restrict__ agg,
                          float* __restrict__ cnt, int E) {
  if (flags[iter] == 0) return;  // converged: no-op iteration
  int gid = blockIdx.x * blockDim.x + threadIdx.x;
  int e = gid >> 5;
  if (e >= E) return;
  int j = gid & 31;
  int src = ei[e];
  if (!has_data[src]) return;    // valid==0 edges contribute nothing
  int dst = ei[E + e];
  float2 v = *(const float2*)(xw + (size_t)src * 64 + j * 2);
  float* ap = agg + (size_t)dst * 64 + j * 2;
  atomicAdd(ap + 0, v.x);
  atomicAdd(ap + 1, v.y);
  if (j == 0) atomicAdd(cnt + dst, 1.0f);
}

// one wave per node: fill zeros with mean, recompute has_data, zero agg/cnt for
// next round, set flags[iter+1] if zeros remain
__global__ void k_update(int* __restrict__ flags, int iter,
                         float* __restrict__ xw, float* __restrict__ agg,
                         float* __restrict__ cnt, int* __restrict__ has_data, int N) {
  if (flags[iter] == 0) return;
  int tid = blockIdx.x * blockDim.x + threadIdx.x;
  int wave = tid >> 5, lane = tid & 31;
  if (wave >= N) return;
  float c = cnt[wave];
  float inv = 1.f / (c > 0.f ? c : 1.f);
  size_t off = (size_t)wave * 64 + lane * 2;
  float2 xv = *(const float2*)(xw + off);
  float2 av = *(const float2*)(agg + off);
  float nx = (xv.x == 0.f) ? av.x * inv : xv.x;
  float ny = (xv.y == 0.f) ? av.y * inv : xv.y;
  float2 nv = {nx, ny};
  *(float2*)(xw + off) = nv;
  float2 z = {0.f, 0.f};
  *(float2*)(agg + off) = z;
  int anynz = __any((nx != 0.f) || (ny != 0.f));
  int anyz  = __any((nx == 0.f) || (ny == 0.f));
  if (lane == 0) {
    cnt[wave] = 0.f;
    has_data[wave] = anynz ? 1 : 0;
    if (anyz) atomicOr(flags + iter + 1, 1);
  }
}

// ---------------------------------------------------------------------------
// LSTM via V_WMMA_F32_16X16X4_F32: D[gate j][node n] = A(W,16x4) x B(data,4x16) + C(bias)
//   A lanes 0-15 : row j=lane, k=0..1 -> w_ih[j][0..1]  (j>=8 -> 0)
//   A lanes 16-31: row j=lane-16, k=2..3 -> w_hh[j][0..1]
//   B lanes 0-15 : k=0,1 = x0,x1 of node (lane)   | lanes 16-31: k=2,3 = h0,h1
//   C lanes 0-15 : c[r] = b_ih[r]+b_hh[r]         | lanes 16-31: 0
//   D lane n (<16) holds gates 0..7 of node n in d[0..7] -> in-lane activations.
__global__ void k_lstm(const float* __restrict__ xw,
                       const float* __restrict__ w_ih, const float* __restrict__ w_hh,
                       const float* __restrict__ b_ih, const float* __restrict__ b_hh,
                       float* __restrict__ out, int N, int T) {
  int tid = blockIdx.x * blockDim.x + threadIdx.x;
  int wave = tid >> 5;
  int lane = tid & 31;
  int base = wave * 16;
  if (base >= N) return;          // wave-uniform: EXEC stays all-1 for live waves
  int nl = lane & 15;
  bool isLo = lane < 16;
  int node = base + nl;
  int nodeC = node < N ? node : (N - 1);
  bool doStore = isLo && (node < N);

  v2f A;
  A[0] = 0.f; A[1] = 0.f;
  if (nl < 8) {
    const float* w = isLo ? w_ih : w_hh;
    A[0] = w[nl * 2 + 0];
    A[1] = w[nl * 2 + 1];
  }

  v8f C;
#pragma unroll
  for (int r = 0; r < 8; ++r) {
    float bb = b_ih[r] + b_hh[r];
    C[r] = isLo ? bb : 0.f;
  }

  float c0 = 0.f, c1 = 0.f;       // cell state (meaningful in lanes 0-15)
  float h0s = 0.f, h1s = 0.f;     // h broadcast into lanes 16-31

  const float* xwp = xw + (size_t)nodeC * 64;

  for (int t = 0; t < T; ++t) {
    float2 xv = *(const float2*)(xwp + t * 2);
    v2f B;
    B[0] = isLo ? xv.x : h0s;
    B[1] = isLo ? xv.y : h1s;
    v8f D = __builtin_amdgcn_wmma_f32_16x16x4_f32(
        /*neg_a=*/false, A, /*neg_b=*/false, B,
        /*c_mod=*/(short)0, C, /*reuse_a=*/false, /*reuse_b=*/false);
    float i0 = sigm(D[0]), i1 = sigm(D[1]);
    float f0 = sigm(D[2]), f1 = sigm(D[3]);
    float g0 = tanh_fast(D[4]), g1 = tanh_fast(D[5]);
    float o0 = sigm(D[6]), o1 = sigm(D[7]);
    c0 = f0 * c0 + i0 * g0;
    c1 = f1 * c1 + i1 * g1;
    float h0 = o0 * tanh_fast(c0);
    float h1 = o1 * tanh_fast(c1);
    if (doStore) {
      float2 hv; hv.x = h0; hv.y = h1;
      *(float2*)(out + ((size_t)t * N + node) * 2) = hv;  // coalesced [T][N][F]
    }
    h0s = __shfl(h0, nl, 32);     // lanes 16-31 fetch h from lane-16
    h1s = __shfl(h1, nl, 32);
  }
}

// ---------------------------------------------------------------------------
extern "C" void kernel_launch(void* const* d_in, const int* in_sizes, int n_in,
                              void* d_out, int out_size, void* d_ws, size_t ws_size,
                              hipStream_t stream) {
  const float* x = (const float*)d_in[0];
  const int* ei = (const int*)d_in[1];
  const unsigned char* mask = (const unsigned char*)d_in[2];  // jnp bool -> 1 byte
  const float* w_ih = (const float*)d_in[6];
  const float* w_hh = (const float*)d_in[7];
  const float* b_ih = (const float*)d_in[8];
  const float* b_hh = (const float*)d_in[9];
  float* out = (float*)d_out;

  int N = in_sizes[2];
  int E = in_sizes[1] / 2;
  int T = in_sizes[0] / (N * 2);

  char* ws = (char*)d_ws;
  size_t off = 0;
  int* flags = (int*)(ws + off);      off = ws_align(off + 32 * sizeof(int));
  int* has_data = (int*)(ws + off);   off = ws_align(off + (size_t)N * sizeof(int));
  float* cnt = (float*)(ws + off);    off = ws_align(off + (size_t)N * sizeof(float));
  float* xw = (float*)(ws + off);     off = ws_align(off + (size_t)N * 64 * sizeof(float));
  float* agg = out;  // reuse d_out as accumulator; LSTM fully overwrites it last

  hipLaunchKernelGGL(k_init_flags, dim3(1), dim3(32), 0, stream, flags);

  dim3 mgrid((N + 255) / 256, (unsigned)T);
  hipLaunchKernelGGL(k_mask_relayout, mgrid, dim3(256), 0, stream, x, mask, xw, N);

  int nodeBlocks = (int)(((long long)N * 32 + 255) / 256);
  hipLaunchKernelGGL(k_prep0, dim3(nodeBlocks), dim3(256), 0, stream,
                     xw, agg, cnt, has_data, flags, N);

  int edgeBlocks = (int)(((long long)E * 32 + 255) / 256);
  for (int it = 0; it < 20; ++it) {  // reference caps at 20; converged rounds no-op
    hipLaunchKernelGGL(k_scatter, dim3(edgeBlocks), dim3(256), 0, stream,
                       flags, it, ei, has_data, xw, agg, cnt, E);
    hipLaunchKernelGGL(k_update, dim3(nodeBlocks), dim3(256), 0, stream,
                       flags, it, xw, agg, cnt, has_data, N);
  }

  int lstmWaves = (N + 15) / 16;
  int lstmBlocks = (int)(((long long)lstmWaves * 32 + 255) / 256);
  hipLaunchKernelGGL(k_lstm, dim3(lstmBlocks), dim3(256), 0, stream,
                     xw, w_ih, w_hh, b_ih, b_hh, out, N, T);
}